// GradientHistLoss_85598698209787
// MI455X (gfx1250) — compile-verified
//
#include <hip/hip_runtime.h>
#include <hip/hip_bf16.h>

typedef __attribute__((ext_vector_type(2))) float v2f;
typedef __attribute__((ext_vector_type(8))) float v8f;

#define NPER   294912      // 2*384*384 elements per sample
#define NSAMP  4
#define CHUNKS 72          // 72 * 4096 == NPER
#define RANK0  280165u     // floor(0.95f * (NPER-1))

// ---------------- ws layout (32-bit words) ----------------
// [0    ,8192) radix counts [round][sample][rank][256] (u32)
// [8192 ,8200) prefix       [sample][rank]             (u32)
// [8200 ,8208) krem         [sample][rank]             (u32)
// [8208 ,8216) selected val [sample][rank]             (f32)
// [8216 ,8220) maxval       [sample]                   (f32)
// [8224 ,8736) histograms   [sample][{p,g}][64]        (f32)

__global__ void init_ws_k(unsigned* __restrict__ ws) {
  int i = blockIdx.x * 256 + threadIdx.x;
  if (i >= 8736) return;
  if (i >= 8200 && i < 8208) ws[i] = RANK0 + (unsigned)((i - 8200) & 1);
  else                       ws[i] = 0u;
}

// counts current-byte occupancy among elements matching the fixed prefix
__global__ void radix_hist_k(const float* __restrict__ gt,
                             unsigned* __restrict__ countsR,
                             const unsigned* __restrict__ prefix,
                             int shift) {
  __shared__ unsigned sc[256];
  int tid = threadIdx.x;
  int s = blockIdx.y, r = blockIdx.z;
  sc[tid] = 0u;
  __syncthreads();
  unsigned pmask = (shift == 24) ? 0u : (0xFFFFFFFFu << (shift + 8));
  unsigned pref  = prefix[s * 2 + r];
  const float4* g4 = reinterpret_cast<const float4*>(
      gt + (size_t)s * NPER + (size_t)blockIdx.x * 4096);
  for (int i = 0; i < 4; ++i) {
    float4 v = g4[i * 256 + tid];
    unsigned u;
    u = __float_as_uint(v.x); if ((u & pmask) == pref) atomicAdd(&sc[(u >> shift) & 255u], 1u);
    u = __float_as_uint(v.y); if ((u & pmask) == pref) atomicAdd(&sc[(u >> shift) & 255u], 1u);
    u = __float_as_uint(v.z); if ((u & pmask) == pref) atomicAdd(&sc[(u >> shift) & 255u], 1u);
    u = __float_as_uint(v.w); if ((u & pmask) == pref) atomicAdd(&sc[(u >> shift) & 255u], 1u);
  }
  __syncthreads();
  atomicAdd(&countsR[(s * 2 + r) * 256 + tid], sc[tid]);
}

// find bucket containing the remaining rank; narrow prefix
__global__ void radix_scan_k(const unsigned* __restrict__ countsR,
                             unsigned* __restrict__ prefix,
                             unsigned* __restrict__ krem,
                             float* __restrict__ val,
                             int shift) {
  __shared__ unsigned sc[256];
  int tid = threadIdx.x;
  int sr = blockIdx.x;  // sample*2 + rank
  sc[tid] = countsR[sr * 256 + tid];
  __syncthreads();
  if (tid == 0) {
    unsigned k = krem[sr], cum = 0;
    int bsel = 255; bool found = false;
    for (int b = 0; b < 256; ++b) {
      unsigned cb = sc[b];
      if (k < cum + cb) { bsel = b; found = true; break; }
      cum += cb;
    }
    if (!found) { bsel = 255; cum -= sc[255]; }
    krem[sr] = k - cum;
    unsigned np = prefix[sr] | ((unsigned)bsel << shift);
    prefix[sr] = np;
    if (shift == 0) val[sr] = __uint_as_float(np);
  }
}

// linear-interpolated 95th percentile, frac computed in f32 like jax
__global__ void qmax_k(const float* __restrict__ val, float* __restrict__ maxval) {
  int s = threadIdx.x;
  if (s < NSAMP) {
    float pos = 0.95f * (float)(NPER - 1);
    float fr  = pos - floorf(pos);
    float v0 = val[2 * s], v1 = val[2 * s + 1];
    maxval[s] = v0 + fr * (v1 - v0);
  }
}

__device__ __forceinline__ void scatter_tri(float* h, float t) {
  float fj = floorf(t);
  int   j  = (int)fj;
  float f  = t - fj;
  if (j  >= 0 && j  < 64) atomicAdd(&h[j],     1.0f - f);   // ds_add_f32
  int j1 = j + 1;
  if (j1 >= 0 && j1 < 64) atomicAdd(&h[j1],    f);
}

__global__ void soft_hist_k(const float* __restrict__ pred,
                            const float* __restrict__ gt,
                            const float* __restrict__ maxval,
                            float* __restrict__ hist) {
  __shared__ float sp[64], sg[64];
  int tid = threadIdx.x;
  int s = blockIdx.y;
  if (tid < 64)       sp[tid]      = 0.0f;
  else if (tid < 128) sg[tid - 64] = 0.0f;
  __syncthreads();
  float inv = 64.0f / maxval[s];  // 1 / bin_width
  size_t base = (size_t)s * NPER + (size_t)blockIdx.x * 4096;
  const float4* p4 = reinterpret_cast<const float4*>(pred + base);
  const float4* g4 = reinterpret_cast<const float4*>(gt + base);
  for (int i = 0; i < 4; ++i) {
    float4 p = p4[i * 256 + tid];
    float4 g = g4[i * 256 + tid];
    scatter_tri(sp, p.x * inv); scatter_tri(sp, p.y * inv);
    scatter_tri(sp, p.z * inv); scatter_tri(sp, p.w * inv);
    scatter_tri(sg, g.x * inv); scatter_tri(sg, g.y * inv);
    scatter_tri(sg, g.z * inv); scatter_tri(sg, g.w * inv);
  }
  __syncthreads();
  if (tid < 64)       atomicAdd(&hist[s * 128 + tid], sp[tid]);
  else if (tid < 128) atomicAdd(&hist[s * 128 + tid], sg[tid - 64]);
}

// Sum 64 values laid out as a 16x4 f32 A-matrix using the matrix pipe:
// D = A x ones(4x16) gives row sums; combine rows via lanes 0 and 16.
__device__ __forceinline__ float wmma_sum64(float x0, float x1) {
  v2f a; a[0] = x0;   a[1] = x1;
  v2f b; b[0] = 1.0f; b[1] = 1.0f;
  v8f c = {};
  c = __builtin_amdgcn_wmma_f32_16x16x4_f32(false, a, false, b,
                                            (short)0, c, false, false);
  float srow = c[0] + c[1] + c[2] + c[3] + c[4] + c[5] + c[6] + c[7];
  return __shfl(srow, 0, 32) + __shfl(srow, 16, 32);
}

__global__ void finalize_k(const float* __restrict__ hist, float* __restrict__ out) {
  int lane = threadIdx.x;              // exactly one wave32, EXEC all 1s
  int m  = lane & 15;
  int i0 = m * 4 + ((lane >> 4) << 1); // A-matrix (16x4) element mapping
  int i1 = i0 + 1;
  float w0 = expf(0.4f * (float)i0 * (1.0f / 64.0f));
  float w1 = expf(0.4f * (float)i1 * (1.0f / 64.0f));
  float acc = 0.0f;
  for (int s = 0; s < NSAMP; ++s) {
    const float* ph = hist + s * 128;
    const float* gh = ph + 64;
    float p0 = ph[i0], p1 = ph[i1];
    float g0 = gh[i0], g1 = gh[i1];
    float Sp = wmma_sum64(p0, p1);
    float Sg = wmma_sum64(g0, g1);
    float rp = 1.0f / Sp, rg = 1.0f / Sg;
    float t0 = fabsf(p0 * rp - g0 * rg) * w0;
    float t1 = fabsf(p1 * rp - g1 * rg) * w1;
    acc += wmma_sum64(t0, t1) * (1.0f / 64.0f);
  }
  if (lane == 0) out[0] = acc * 0.25f;
}

extern "C" void kernel_launch(void* const* d_in, const int* in_sizes, int n_in,
                              void* d_out, int out_size, void* d_ws, size_t ws_size,
                              hipStream_t stream) {
  (void)in_sizes; (void)n_in; (void)out_size; (void)ws_size;
  const float* pred = (const float*)d_in[0];
  const float* gt   = (const float*)d_in[1];
  unsigned* wsu    = (unsigned*)d_ws;
  unsigned* counts = wsu;
  unsigned* prefix = wsu + 8192;
  unsigned* krem   = wsu + 8200;
  float*    val    = (float*)(wsu + 8208);
  float*    maxval = (float*)(wsu + 8216);
  float*    hist   = (float*)(wsu + 8224);

  init_ws_k<<<35, 256, 0, stream>>>(wsu);

  for (int round = 0; round < 4; ++round) {
    int shift = 24 - 8 * round;
    unsigned* cR = counts + round * 2048;  // 8 selections * 256 bins
    radix_hist_k<<<dim3(CHUNKS, NSAMP, 2), 256, 0, stream>>>(gt, cR, prefix, shift);
    radix_scan_k<<<8, 256, 0, stream>>>(cR, prefix, krem, val, shift);
  }

  qmax_k<<<1, 32, 0, stream>>>(val, maxval);
  soft_hist_k<<<dim3(CHUNKS, NSAMP), 256, 0, stream>>>(pred, gt, maxval, hist);
  finalize_k<<<1, 32, 0, stream>>>(hist, (float*)d_out);
}